// NaiveCodebook_47536698032991
// MI455X (gfx1250) — compile-verified
//
#include <hip/hip_runtime.h>
#include <hip/hip_bf16.h>
#include <math.h>

// Problem constants (from reference): B=16, C_IN=64, E=H=W=N=128
#define NB   16
#define CIN  64
#define EE   128
#define HW   128

typedef __attribute__((ext_vector_type(2))) float v2f;
typedef __attribute__((ext_vector_type(8))) float v8f;

static __device__ __forceinline__ v8f wmma4(v2f a, v2f b, v8f c) {
  // D = A(16x4 f32) * B(4x16 f32) + C(16x16 f32)  ->  v_wmma_f32_16x16x4_f32
  return __builtin_amdgcn_wmma_f32_16x16x4_f32(
      /*neg_a=*/false, a, /*neg_b=*/false, b,
      /*c_mod=*/(short)0, c, /*reuse_a=*/false, /*reuse_b=*/false);
}

// ---------------- CDNA5 async / TDM feature detection --------------------
#if defined(__gfx1250__) && __has_builtin(__builtin_amdgcn_global_load_async_to_lds_b128)
#define HAVE_ASYNC_LDS 1
#else
#define HAVE_ASYNC_LDS 0
#endif

#if defined(__gfx1250__) && __has_builtin(__builtin_amdgcn_tensor_load_to_lds) && \
    __has_builtin(__builtin_amdgcn_s_wait_tensorcnt)
#define HAVE_TDM 1
#else
#define HAVE_TDM 0
#endif

// G->LDS 16B async copy (ASYNCcnt path); falls back to a plain copy.
// Builtin prototype (from hipcc diagnostic): (v4i AS1*, v4i AS3*, imm, imm)
// with v4i = int __attribute__((vector_size(16))).
static __device__ __forceinline__ void g2l_b128(const float* g, float* l) {
#if HAVE_ASYNC_LDS
  typedef int gv4i __attribute__((vector_size(16)));
  __builtin_amdgcn_global_load_async_to_lds_b128(
      (__attribute__((address_space(1))) gv4i*)g,
      (__attribute__((address_space(3))) gv4i*)l, 0, 0);
#else
  *(float4*)l = *(const float4*)g;
#endif
}

static __device__ __forceinline__ void wait_async_lds() {
#if HAVE_ASYNC_LDS
#if __has_builtin(__builtin_amdgcn_s_wait_asynccnt)
  __builtin_amdgcn_s_wait_asynccnt(0);
#else
  asm volatile("s_wait_asynccnt 0x0" ::: "memory");
#endif
#endif
}

#if HAVE_TDM
typedef __attribute__((ext_vector_type(4))) unsigned int u32x4;
typedef __attribute__((ext_vector_type(4))) int i32x4;
typedef __attribute__((ext_vector_type(8))) int i32x8;

// One 2-D tile: dim1 rows of dim0 fp32 elements, global row stride = stride0
// elements, dense into LDS at lds_byte. D# packed per CDNA5 ISA 8.3/8.4.
static __device__ __forceinline__ void tdm_load_2d(const float* g, unsigned lds_byte,
                                                   unsigned dim0, unsigned dim1,
                                                   unsigned stride0) {
  unsigned long long ga = (unsigned long long)g;
  u32x4 g0;
  g0[0] = 1u;                                        // count=1 (valid user D#)
  g0[1] = lds_byte;                                  // lds_addr
  g0[2] = (unsigned)(ga & 0xFFFFFFFFull);            // global_addr[31:0]
  g0[3] = (unsigned)((ga >> 32) & 0x01FFFFFFull)     // global_addr[56:32]
          | (2u << 30);                              // type=2 ("image")
  i32x8 g1;
  g1[0] = (int)(2u << 16);                           // wg_mask=0, data_size=4B
  g1[1] = (int)(dim0 << 16);                         // tensor_dim0[15:0] @63:48
  g1[2] = (int)((dim0 >> 16) | (dim1 << 16));        // dim0[31:16] | dim1[15:0]
  g1[3] = (int)((dim1 >> 16) | (dim0 << 16));        // dim1[31:16] | tile_dim0
  g1[4] = (int)(dim1 & 0xFFFFu);                     // tile_dim1 (tile_dim2=0)
  g1[5] = (int)stride0;                              // dim0_stride[31:0]
  g1[6] = 0;                                         // stride hi / dim1_stride
  g1[7] = 0;
  i32x4 z4 = {0, 0, 0, 0};
#if __clang_major__ >= 23
  i32x8 z8 = {0, 0, 0, 0, 0, 0, 0, 0};
  __builtin_amdgcn_tensor_load_to_lds(g0, g1, z4, z4, z8, 0);
#else
  __builtin_amdgcn_tensor_load_to_lds(g0, g1, z4, z4, 0);
#endif
}
#endif  // HAVE_TDM

// -------------------------------------------------------------------------
// K0a/K0b/K0c: repack w_in / w_out / book into exact WMMA fragment layout:
// float index ((frag)*64 + lane*2 + j)  ->  one b64 load per lane per frag.
// A-fragment lane map: lm=lane&15 (M), kh=lane>>4, K = kk + kh*2 + j.
// -------------------------------------------------------------------------
__global__ __launch_bounds__(256) void k_pack_win(const float* __restrict__ w,
                                                  float* __restrict__ p) {
  int i = blockIdx.x * 256 + threadIdx.x;            // < 73728 exact
  int j = i & 1, lane = (i >> 1) & 31, kk4 = (i >> 6) & 15, r = i >> 10;
  int tap = r % 9, mt = r / 9;
  int lm = lane & 15, kh = lane >> 4;
  p[i] = w[((mt * 16 + lm) * CIN + kk4 * 4 + kh * 2 + j) * 9 + tap];
}

__global__ __launch_bounds__(256) void k_pack_wout(const float* __restrict__ w,
                                                   float* __restrict__ p) {
  int i = blockIdx.x * 256 + threadIdx.x;            // < 73728 exact
  int j = i & 1, lane = (i >> 1) & 31, kk4 = (i >> 6) & 31, r = i >> 11;
  int tap = r % 9, mt = r / 9;
  int lm = lane & 15, kh = lane >> 4;
  p[i] = w[((mt * 16 + lm) * EE + kk4 * 4 + kh * 2 + j) * 9 + tap];
}

__global__ __launch_bounds__(256) void k_pack_book(const float* __restrict__ bk,
                                                   float* __restrict__ p) {
  int i = blockIdx.x * 256 + threadIdx.x;            // < 16384 exact
  int j = i & 1, lane = (i >> 1) & 31, kk4 = (i >> 6) & 31, nt = i >> 11;
  int lm = lane & 15, kh = lane >> 4;
  p[i] = bk[(nt * 16 + lm) * EE + kk4 * 4 + kh * 2 + j];
}

// -------------------------------------------------------------------------
// K1: diff = image_1 - image_2  (conv_in bias cancels; conv is linear)
// -------------------------------------------------------------------------
__global__ __launch_bounds__(256) void k_diff(const float* __restrict__ a,
                                              const float* __restrict__ b,
                                              float* __restrict__ d) {
  int i = blockIdx.x * 256 + threadIdx.x;            // float4 index, exact grid
  float4 va = ((const float4*)a)[i], vb = ((const float4*)b)[i];
  float4 r;
  r.x = va.x - vb.x; r.y = va.y - vb.y; r.z = va.z - vb.z; r.w = va.w - vb.w;
  ((float4*)d)[i] = r;
}

// -------------------------------------------------------------------------
// K2: input_data = conv3x3(diff, w_in)   [B,E,H,W], no bias
// Implicit GEMM: M=E=128 (8 waves x 16), N=16 px/block, K=64x9 taps.
// Slab interior loaded via async b128 G->LDS; halo zero-filled by stores.
// LDS stride 24 floats so the interior (px offset 4) is 16B aligned.
// -------------------------------------------------------------------------
__global__ __launch_bounds__(256) void k_conv_in(const float* __restrict__ din,
                                                 const float* __restrict__ wpk,
                                                 float* __restrict__ xd) {
  __shared__ float s[3 * 64 * 24];                   // 18,432 B
  const int b = blockIdx.z, y = blockIdx.y, x0 = blockIdx.x * 16;
  const int tid = threadIdx.x;

  // halo columns (px index 3 and 20) incl. OOB -> 0
  for (int i = tid; i < 384; i += 256) {
    int row = i / 128, rem = i - row * 128, c = rem >> 1, side = rem & 1;
    int p = side ? 20 : 3;
    int yy = y + row - 1, xx = x0 + p - 4;
    float v = 0.0f;
    if ((unsigned)yy < 128u && (unsigned)xx < 128u)
      v = din[((b * CIN + c) * HW + yy) * HW + xx];
    s[row * 1536 + c * 24 + p] = v;
  }
  // interior 16 px per (row,c): 4 lanes x b128 async DMA
  {
    int c = tid >> 2, q = tid & 3;                   // 256 threads = 64c x 4q
    for (int row = 0; row < 3; ++row) {
      int yy = y + row - 1;
      float* dst = &s[row * 1536 + c * 24 + 4 + q * 4];
      if ((unsigned)yy < 128u) {
        g2l_b128(&din[((b * CIN + c) * HW + yy) * HW + x0 + q * 4], dst);
      } else {
        dst[0] = 0.f; dst[1] = 0.f; dst[2] = 0.f; dst[3] = 0.f;
      }
    }
  }
  wait_async_lds();
  __syncthreads();

  const int lane = tid & 31, wave = tid >> 5;
  const int lm = lane & 15, kh = lane >> 4;

  v8f acc = {};
  const v2f* ap = (const v2f*)wpk;
  __builtin_prefetch(&wpk[(wave * 9) * 16 * 64], 0, 0);   // global_prefetch_b8

  for (int tap = 0; tap < 9; ++tap) {
    const int dy = tap / 3, dx = tap - dy * 3;
    const float* sp = &s[dy * 1536 + (kh * 2) * 24 + lm + dx + 3];
    const int fbase = (wave * 9 + tap) * 16;
#pragma unroll
    for (int kk = 0; kk < 64; kk += 4) {
      v2f a = ap[(fbase + (kk >> 2)) * 32 + lane];   // one b64 per fragment
      v2f bb;
      bb[0] = sp[(kk + 0) * 24];
      bb[1] = sp[(kk + 1) * 24];
      acc = wmma4(a, bb, acc);
    }
  }

#pragma unroll
  for (int r = 0; r < 8; ++r) {
    int e = wave * 16 + kh * 8 + r;                  // C/D layout
    xd[((b * EE + e) * HW + y) * HW + x0 + lm] = acc[r];
  }
}

// -------------------------------------------------------------------------
// K3a: x_sq[b,y,e] = sum_d input_data[b,e,y,d]^2
// -------------------------------------------------------------------------
__global__ __launch_bounds__(128) void k_xsq(const float* __restrict__ xd,
                                             float* __restrict__ xsq) {
  const int y = blockIdx.x, b = blockIdx.y, e = threadIdx.x;
  const float* p = &xd[((b * EE + e) * HW + y) * HW];
  float sacc = 0.0f;
#pragma unroll 4
  for (int d = 0; d < 128; ++d) sacc += p[d] * p[d];
  xsq[(b * HW + y) * EE + e] = sacc;
}

// -------------------------------------------------------------------------
// K3: indices[b,y,n] = argmin_e ( x_sq[e] - 2 * <x[b,e,y,:], book[n,:]> )
// The 128x512B slab (row stride 64KB) is a textbook TDM 2-D tile:
// wave 0 issues one tensor_load_to_lds, drains TENSORcnt, block barriers.
// -------------------------------------------------------------------------
__global__ __launch_bounds__(256) void k_argmin(const float* __restrict__ xd,
                                                const float* __restrict__ bpk,
                                                const float* __restrict__ xsq,
                                                int* __restrict__ idx) {
  __shared__ float sx[128 * 128];                    // [e][d], 64KB
  const int y = blockIdx.x, b = blockIdx.y, tid = threadIdx.x;
  const float* gsrc = &xd[((b * EE) * HW + y) * HW]; // row e stride = 16384 f32

#if HAVE_TDM
  if (tid < 32) {                                    // one wave drives the TDM
    unsigned lds_off =
        (unsigned)(unsigned long long)(__attribute__((address_space(3))) float*)sx;
    tdm_load_2d(gsrc, lds_off, 128u, 128u, 16384u);
    __builtin_amdgcn_s_wait_tensorcnt((unsigned short)0);
  }
  __syncthreads();
#else
  for (int i = tid; i < 4096; i += 256) {            // 4096 x b128
    int e = i >> 5, d4 = (i & 31) << 2;
    g2l_b128(&gsrc[e * 16384 + d4], &sx[e * 128 + d4]);
  }
  wait_async_lds();
  __syncthreads();
#endif

  const int lane = tid & 31, wave = tid >> 5;
  const int lm = lane & 15, kh = lane >> 4;

  const v2f* sxv = (const v2f*)sx;
  const v2f* bpv = (const v2f*)bpk;
  const float* xq = &xsq[(b * HW + y) * EE];

  float best = 3.4e38f;
  int bestE = 0;

  for (int mt = 0; mt < 8; ++mt) {
    v8f acc = {};
    const int abase = (mt * 16 + lm) * 64 + kh;      // ds_load_b64 A-fragments
#pragma unroll 8
    for (int kk = 0; kk < 128; kk += 4) {
      v2f a = sxv[abase + (kk >> 1)];
      v2f bb = bpv[(wave * 32 + (kk >> 2)) * 32 + lane];
      acc = wmma4(a, bb, acc);
    }
#pragma unroll
    for (int r = 0; r < 8; ++r) {
      int e = mt * 16 + kh * 8 + r;                  // ascending e per lane
      float sc = xq[e] - 2.0f * acc[r];
      if (sc < best) { best = sc; bestE = e; }       // strict < => first index
    }
  }

  // lanes L and L+16 hold the same n with disjoint e-halves: combine.
  float ob = __shfl_xor(best, 16, 32);
  int   oe = __shfl_xor(bestE, 16, 32);
  if (ob < best || (ob == best && oe < bestE)) { best = ob; bestE = oe; }
  if (lane < 16) idx[(b * HW + y) * 128 + wave * 16 + lane] = bestE;
}

// -------------------------------------------------------------------------
// K4: ratio[b,j,d] = ||res[:,j,d]|| / ||rv[:,j,d]|| + 1e-6   (sum over axis 1)
// -------------------------------------------------------------------------
__global__ __launch_bounds__(128) void k_norms(const float* __restrict__ xd,
                                               const float* __restrict__ book,
                                               const float* __restrict__ rv,
                                               const int* __restrict__ idx,
                                               float* __restrict__ ratio) {
  const int j = blockIdx.x, b = blockIdx.y, d = threadIdx.x;
  float ar = 0.0f, av = 0.0f;
  for (int i = 0; i < 128; ++i) {
    int id = idx[(b * HW + i) * 128 + j];
    float xv = xd[((b * EE + i) * HW + j) * HW + d];
    float bv = book[id * EE + d];
    float r = xv - bv;
    ar += r * r;
    float rw = rv[((b * EE + i) * HW + j) * HW + d];
    av += rw * rw;
  }
  ratio[(b * HW + j) * EE + d] = sqrtf(ar) / sqrtf(av) + 1e-6f;
}

// -------------------------------------------------------------------------
// K5: quantized = input_data + ratio[b,j,d] * rv   (in place over xd)
// -------------------------------------------------------------------------
__global__ __launch_bounds__(256) void k_quant(float* __restrict__ xd,
                                               const float* __restrict__ rv,
                                               const float* __restrict__ ratio) {
  int v = blockIdx.x * 256 + threadIdx.x;            // float4 index, exact grid
  int base = v * 4;
  int b = base >> 21;
  int j = (base >> 7) & 127;
  int d = base & 127;
  float4 x = *(float4*)&xd[base];
  float4 r = *(const float4*)&rv[base];
  float4 t = *(const float4*)&ratio[(b << 14) + (j << 7) + d];
  x.x += t.x * r.x; x.y += t.y * r.y; x.z += t.z * r.z; x.w += t.w * r.w;
  *(float4*)&xd[base] = x;
}

// -------------------------------------------------------------------------
// K6: out = conv3x3(quantized, w_out) + b_out   [B,64,128,128]
// M=64 (4 tiles) x N=32 px (2 subtiles) per block, K=128x9.
// Slab: LDS stride 40 floats, interior at px offset 4 (16B aligned), async.
// -------------------------------------------------------------------------
__global__ __launch_bounds__(256) void k_conv_out(const float* __restrict__ q,
                                                  const float* __restrict__ wpk,
                                                  const float* __restrict__ bias,
                                                  float* __restrict__ out) {
  __shared__ float s[3 * 128 * 40];                  // 61,440 B
  const int b = blockIdx.z, y = blockIdx.y, x0 = blockIdx.x * 32;
  const int tid = threadIdx.x;

  for (int i = tid; i < 768; i += 256) {             // halo columns px 3 / 36
    int row = i / 256, rem = i - row * 256, c = rem >> 1, side = rem & 1;
    int p = side ? 36 : 3;
    int yy = y + row - 1, xx = x0 + p - 4;
    float v = 0.0f;
    if ((unsigned)yy < 128u && (unsigned)xx < 128u)
      v = q[((b * EE + c) * HW + yy) * HW + xx];
    s[row * 5120 + c * 40 + p] = v;
  }
  for (int row = 0; row < 3; ++row) {                // interior via async b128
    int yy = y + row - 1;
    for (int i = tid; i < 1024; i += 256) {
      int c = i >> 3, qd = i & 7;
      float* dst = &s[row * 5120 + c * 40 + 4 + qd * 4];
      if ((unsigned)yy < 128u) {
        g2l_b128(&q[((b * EE + c) * HW + yy) * HW + x0 + qd * 4], dst);
      } else {
        dst[0] = 0.f; dst[1] = 0.f; dst[2] = 0.f; dst[3] = 0.f;
      }
    }
  }
  wait_async_lds();
  __syncthreads();

  const int lane = tid & 31, wave = tid >> 5;
  const int lm = lane & 15, kh = lane >> 4;
  const int mt = wave & 3, n0 = (wave >> 2) * 16;

  v8f acc = {};
  const v2f* ap = (const v2f*)wpk;
  __builtin_prefetch(&wpk[(mt * 9) * 32 * 64], 0, 0);

  for (int tap = 0; tap < 9; ++tap) {
    const int dy = tap / 3, dx = tap - dy * 3;
    const float* sp = &s[dy * 5120 + (kh * 2) * 40 + n0 + lm + dx + 3];
    const int fbase = (mt * 9 + tap) * 32;
#pragma unroll 8
    for (int kk = 0; kk < 128; kk += 4) {
      v2f a = ap[(fbase + (kk >> 2)) * 32 + lane];
      v2f bb;
      bb[0] = sp[(kk + 0) * 40];
      bb[1] = sp[(kk + 1) * 40];
      acc = wmma4(a, bb, acc);
    }
  }

#pragma unroll
  for (int r = 0; r < 8; ++r) {
    int co = mt * 16 + kh * 8 + r;
    out[((b * CIN + co) * HW + y) * HW + x0 + n0 + lm] = acc[r] + bias[co];
  }
}

// -------------------------------------------------------------------------
extern "C" void kernel_launch(void* const* d_in, const int* in_sizes, int n_in,
                              void* d_out, int out_size, void* d_ws, size_t ws_size,
                              hipStream_t stream) {
  (void)in_sizes; (void)n_in; (void)out_size; (void)ws_size;

  const float* img1  = (const float*)d_in[0];
  const float* img2  = (const float*)d_in[1];
  const float* w_in  = (const float*)d_in[2];
  // d_in[3] = b_in: cancels in x1 - x2 (conv is linear) — intentionally unused
  const float* w_out = (const float*)d_in[4];
  const float* b_out = (const float*)d_in[5];
  const float* book  = (const float*)d_in[6];
  const float* rv    = (const float*)d_in[7];

  float* ws    = (float*)d_ws;
  float* diff  = ws;                        // 16,777,216 f32  (67 MB)
  float* xd    = diff + 16777216;           // 33,554,432 f32  (134 MB)
  float* xsq   = xd + 33554432;             //    262,144 f32  (1 MB)
  float* ratio = xsq + 262144;              //    262,144 f32  (1 MB)
  float* pin   = ratio + 262144;            //     73,728 f32
  float* pout  = pin + 73728;               //     73,728 f32
  float* pbook = pout + 73728;              //     16,384 f32

  float* outp = (float*)d_out;              // [16,64,128,128] f32
  int*   idxp = (int*)(outp + 16777216);    // [16,128,128] i32

  k_pack_win <<<288, 256, 0, stream>>>(w_in, pin);
  k_pack_wout<<<288, 256, 0, stream>>>(w_out, pout);
  k_pack_book<<<64, 256, 0, stream>>>(book, pbook);
  k_diff     <<<16384, 256, 0, stream>>>(img1, img2, diff);
  k_conv_in  <<<dim3(8, 128, 16), 256, 0, stream>>>(diff, pin, xd);
  k_xsq      <<<dim3(128, 16), 128, 0, stream>>>(xd, xsq);
  k_argmin   <<<dim3(128, 16), 256, 0, stream>>>(xd, pbook, xsq, idxp);
  k_norms    <<<dim3(128, 16), 128, 0, stream>>>(xd, book, rv, idxp, ratio);
  k_quant    <<<32768, 256, 0, stream>>>(xd, rv, ratio);
  k_conv_out <<<dim3(4, 128, 16), 256, 0, stream>>>(xd, pout, b_out, outp);
}